// DeformableParts_19980187861500
// MI455X (gfx1250) — compile-verified
//
#include <hip/hip_runtime.h>
#include <hip/hip_bf16.h>
#include <math.h>

// ---------------------------------------------------------------------------
// DeformableParts head for MI455X (gfx1250, wave32).
// Convs = implicit GEMM on V_WMMA_F32_16X16X4_F32 (exact fp32).
// Inputs zero-padded AND channel-pair interleaved so each lane's WMMA K-pair
// is one aligned global_load_b64 -> no operand-assembly movs, no predication.
// ---------------------------------------------------------------------------

typedef __attribute__((ext_vector_type(2))) float v2f;
typedef __attribute__((ext_vector_type(8))) float v8f;

#define CIN       128
#define BATCH     2
#define PTOT      13640     // sum of H*W over 5 levels
#define PTOT_POS  3408      // sum of (H/2)*(W/2) over 5 levels
#define TWOPI     6.28318530717958647692f
#define MAXCHP    (82*130)  // padded per-channel extent of the largest level

// ----------------------------------------------------------------------------
// 3x3 SAME conv, Cin=128 -> Cout, via WMMA f32 16x16x4.
//   input : interleaved padded  [B][64 pair][CHP][2]
//   weight: repacked            [Cout][64 pair][9 tap][2]
// One wave per (16 out-ch) x (16 pixel) tile; 4 independent waves per block.
// K = 9*128 = 1152 in steps of 4 (288 WMMAs). Lane feeds
//   A[M=lane&15][K=2*(lane>>4)+{0,1}],  B[K=same][N=lane&15]
// and both K-pairs are single b64 loads. D rows r + 8*(lane>>4), col lane&15.
// mode: 0 = acc+bias ; 2 = exp(scale[level]^2 * (acc+bias))  (box decode)
// ----------------------------------------------------------------------------
template<int H, int W>
__global__ __launch_bounds__(128)
void conv3x3_wmma(const float* __restrict__ in2,  // interleaved padded input
                  const float* __restrict__ w2,   // repacked weights
                  const float* __restrict__ bias,
                  float* __restrict__ out, int out_stride, int out_off,
                  const float* __restrict__ scales, int level, int mode,
                  int Cout)
{
    constexpr int HW  = H * W;
    constexpr int WP  = W + 2;
    constexpr int CHP = (H + 2) * (W + 2);    // padded per-channel pixels
    constexpr int Ntiles = (HW + 15) >> 4;
    const int Mtiles = (Cout + 15) >> 4;

    int t = blockIdx.x * 4 + (threadIdx.x >> 5);     // global wave index
    if (t >= Mtiles * Ntiles * BATCH) return;
    const int mt = t % Mtiles;  t /= Mtiles;
    const int nt = t % Ntiles;
    const int b  = t / Ntiles;

    const int lane = threadIdx.x & 31;
    const int n    = lane & 15;
    const int kh   = lane >> 4;                 // K-half: pair index offset 0/1

    const int m  = (mt << 4) + n;               // A row this lane feeds
    const int mA = (m < Cout) ? m : (Cout - 1); // clamp (unused rows unstored)
    const int p  = (nt << 4) + n;               // B pixel this lane feeds
    const int pe = (p < HW) ? p : (HW - 1);     // clamp (garbage cols unstored)
    const int py = pe / W;
    const int px = pe % W;

    // lane base pointers: pair index starts at kh
    const float* __restrict__ aP = w2  + ((size_t)mA * 64 + kh) * 9 * 2;
    const float* __restrict__ bP = in2 + (((size_t)b * 64 + kh) * CHP) * 2;

    v8f acc = {0.f, 0.f, 0.f, 0.f, 0.f, 0.f, 0.f, 0.f};

    #pragma unroll
    for (int tap = 0; tap < 9; ++tap) {
        const int ky = tap / 3, kx = tap % 3;
        const float* __restrict__ xp = bP + ((py + ky) * WP + (px + kx)) * 2;
        const float* __restrict__ wp = aP + tap * 2;
        #pragma unroll 8
        for (int c = 0; c < CIN; c += 4) {      // pair index advances by 2
            v2f a  = *(const v2f*)(wp + (size_t)(c >> 1) * 18);        // 9*2
            v2f bb = *(const v2f*)(xp + (size_t)(c >> 1) * (CHP * 2));
            acc = __builtin_amdgcn_wmma_f32_16x16x4_f32(
                      false, a, false, bb, (short)0, acc, false, false);
        }
    }

    if (p < HW) {
        const int rbase = (lane >> 4) << 3;     // lanes 16-31 hold M = r + 8
        float s2 = 1.f;
        if (mode == 2) { const float s = scales[level]; s2 = s * s; }
        #pragma unroll
        for (int r = 0; r < 8; ++r) {
            const int mo = (mt << 4) + rbase + r;
            if (mo < Cout) {
                float v = acc[r] + bias[mo];
                if (mode == 2) v = expf(s2 * v);
                out[((size_t)b * Cout + mo) * out_stride + out_off + p] = v;
            }
        }
    }
}

// ----------------------------------------------------------------------------
// Repack OIHW weights [O][128][3][3] -> [O][64 pair][9 tap][2].
// Works for any O (multiple conv layers treated as one big O range).
// ----------------------------------------------------------------------------
__global__ __launch_bounds__(256)
void repack_weights(const float* __restrict__ src, float* __restrict__ dst,
                    int total)
{
    const int idx = blockIdx.x * 256 + threadIdx.x;
    if (idx >= total) return;
    const int half = idx & 1;
    int r = idx >> 1;
    const int tap = r % 9;  r /= 9;
    const int pc  = r % 64;
    const int o   = r / 64;
    dst[idx] = src[((size_t)o * 128 + 2 * pc + half) * 9 + tap];
}

// ----------------------------------------------------------------------------
// Copy [B][128][H][W] feature map into zero-padded, pair-interleaved
// [B][64][H+2][W+2][2]. Every cell written each call.
// ----------------------------------------------------------------------------
__global__ __launch_bounds__(256)
void pad_input(const float* __restrict__ src, float* __restrict__ dst,
               int H, int W, int total)
{
    const int idx = blockIdx.x * 256 + threadIdx.x;
    if (idx >= total) return;
    const int WP = W + 2, HP = H + 2, CHP = HP * WP;
    const int half = idx & 1;
    int r = idx >> 1;
    const int pe = r % CHP;  r /= CHP;
    const int pc = r % 64;
    const int b  = r / 64;
    const int yy = pe / WP, xx = pe % WP;
    float v = 0.f;
    if (yy > 0 && yy < HP - 1 && xx > 0 && xx < WP - 1) {
        const int ch = 2 * pc + half;
        v = src[((size_t)b * CIN + ch) * H * W + (yy - 1) * W + (xx - 1)];
    }
    dst[idx] = v;
}

// ----------------------------------------------------------------------------
// GroupNorm(32 groups, 4 ch/group) + ReLU; reads unpadded conv output
// [B][128][HW], writes zero-padded interleaved [B][64][CHP][2].
// ----------------------------------------------------------------------------
__global__ __launch_bounds__(256)
void groupnorm_relu_pad(const float* __restrict__ in,
                        float* __restrict__ out,
                        const float* __restrict__ gamma,
                        const float* __restrict__ beta,
                        int H, int W)
{
    __shared__ float s0[256];
    __shared__ float s1[256];

    const int g   = blockIdx.x & 31;
    const int b   = blockIdx.x >> 5;
    const int tid = threadIdx.x;
    const int HW  = H * W;
    const int N   = 4 * HW;
    const int WP = W + 2, HP = H + 2, CHP = HP * WP;

    const float* __restrict__ base = in + ((size_t)b * CIN + g * 4) * HW;

    float sum = 0.f, sq = 0.f;
    for (int i = tid; i < N; i += 256) {
        const float v = base[i];
        sum += v;
        sq  += v * v;
    }
    s0[tid] = sum; s1[tid] = sq;
    __syncthreads();
    for (int s = 128; s > 0; s >>= 1) {
        if (tid < s) { s0[tid] += s0[tid + s]; s1[tid] += s1[tid + s]; }
        __syncthreads();
    }
    const float mu  = s0[0] / (float)N;
    const float var = s1[0] / (float)N - mu * mu;
    const float inv = rsqrtf(var + 1e-5f);

    // group g covers channels g*4 .. g*4+3 == pairs g*2, g*2+1
    float* __restrict__ ob = out + (((size_t)b * 64 + g * 2) * CHP) * 2;
    const int NP = 4 * CHP;           // 4 channels * padded pixels
    for (int i = tid; i < NP; i += 256) {
        const int cl = i / CHP;       // 0..3 within group
        const int pe = i % CHP;
        const int yy = pe / WP, xx = pe % WP;
        float v = 0.f;
        if (yy > 0 && yy < HP - 1 && xx > 0 && xx < WP - 1) {
            const int   c = g * 4 + cl;
            const float x = base[(size_t)cl * HW + (yy - 1) * W + (xx - 1)];
            v = (x - mu) * inv * gamma[c] + beta[c];
            v = v > 0.f ? v : 0.f;
        }
        // interleaved dst: pair cl>>1, half cl&1
        ob[((size_t)(cl >> 1) * CHP + pe) * 2 + (cl & 1)] = v;
    }
}

// ----------------------------------------------------------------------------
// Positional embedding (sin/cos over cumsum'd mask + maxpooled boxes).
// One thread per (b, y, x, channel), channel in [0,256).
// ----------------------------------------------------------------------------
__global__ __launch_bounds__(256)
void pos_embed_kernel(const unsigned char* __restrict__ mask,
                      const float* __restrict__ boxes,   // d_out boxes base
                      float* __restrict__ pos,           // d_out pos base
                      int H, int W, int h, int w, int box_off, int pos_off)
{
    int idx = blockIdx.x * 256 + threadIdx.x;
    const int c = idx & 255;  idx >>= 8;
    const int x = idx % w;    idx /= w;
    const int y = idx % h;
    const int b = idx / h;
    if (b >= BATCH) return;

    const unsigned char* __restrict__ mb = mask + (size_t)b * h * w;

    float val;
    if (c < 128) {
        float cum = 0.f, last = 0.f;
        if (c < 64) {                 // cumsum down rows at column x
            for (int r = 0; r < h; ++r) {
                const float nm = mb[r * w + x] ? 0.f : 1.f;
                last += nm;
                if (r <= y) cum += nm;
            }
        } else {                      // cumsum across cols at row y
            for (int cc = 0; cc < w; ++cc) {
                const float nm = mb[y * w + cc] ? 0.f : 1.f;
                last += nm;
                if (cc <= x) cum += nm;
            }
        }
        const float e = cum / (last + 1e-6f) * TWOPI;
        const int   i = c & 63;
        const float d = powf(10000.f, (2.f * (float)(i >> 1)) / 64.f);
        const float t = e / d;
        val = (i & 1) ? cosf(t) : sinf(t);
    } else {
        const int i  = c - 128;
        const int pi = i >> 5;        // which of the 4 box channels
        const int j  = i & 31;
        const float* __restrict__ bb =
            boxes + ((size_t)b * 4 + pi) * PTOT + box_off;
        const int Y = 2 * y, X = 2 * x;
        const float m0 = bb[Y * W + X];
        const float m1 = bb[Y * W + X + 1];
        const float m2 = bb[(Y + 1) * W + X];
        const float m3 = bb[(Y + 1) * W + X + 1];
        const float pp = fmaxf(fmaxf(m0, m1), fmaxf(m2, m3));
        const float d  = powf(10000.f, (2.f * (float)(j >> 1)) / 32.f);
        const float t  = pp / d;
        val = (j & 1) ? cosf(t) : sinf(t);
    }
    pos[((size_t)b * 256 + c) * PTOT_POS + pos_off + (y * w + x)] = val;
}

// ----------------------------------------------------------------------------
struct Params {
    const float *scales;
    const float *cls_b, *cls_g, *cls_beta;
    const float *box_b, *box_g, *box_beta;
    const float *logits_b, *boxes_b, *conf_b;
    float *logits_o, *boxes_o, *conf_o, *pos_o;
    float *P0, *P1, *A;                         // padded feat/act, conv out
    float *cls2, *box2, *logits2, *boxes2, *conf2;   // repacked weights
};

template<int H, int W>
static void run_level(const float* feat, const unsigned char* mask,
                      int l, int off_main, int off_pos,
                      const Params& P, hipStream_t stream)
{
    constexpr int HW = H * W;
    constexpr int Nt = (HW + 15) / 16;
    const int w128 = 8 * Nt * BATCH;            // waves for Cout=128
    const int w80  = 5 * Nt * BATCH;
    const int w4   = 1 * Nt * BATCH;
    const dim3 cblk(128), gblk(256), ggrid(BATCH * 32);
    #define CG(nw) dim3(((nw) + 3) / 4)
    const size_t WSTRIDE = (size_t)CIN * CIN * 9;   // layer stride (repacked too)

    const int padTotal = BATCH * CIN * (H + 2) * (W + 2);
    pad_input<<<dim3((padTotal + 255) / 256), gblk, 0, stream>>>(
        feat, P.P0, H, W, padTotal);

    // ---- classification tower ----
    conv3x3_wmma<H, W><<<CG(w128), cblk, 0, stream>>>(
        P.P0, P.cls2, P.cls_b, P.A, HW, 0, P.scales, l, 0, 128);
    groupnorm_relu_pad<<<ggrid, gblk, 0, stream>>>(P.A, P.P1, P.cls_g, P.cls_beta, H, W);
    conv3x3_wmma<H, W><<<CG(w128), cblk, 0, stream>>>(
        P.P1, P.cls2 + WSTRIDE, P.cls_b + CIN, P.A, HW, 0, P.scales, l, 0, 128);
    groupnorm_relu_pad<<<ggrid, gblk, 0, stream>>>(P.A, P.P1, P.cls_g + CIN, P.cls_beta + CIN, H, W);
    conv3x3_wmma<H, W><<<CG(w80), cblk, 0, stream>>>(
        P.P1, P.logits2, P.logits_b, P.logits_o, PTOT, off_main, P.scales, l, 0, 80);

    // ---- box tower ----
    conv3x3_wmma<H, W><<<CG(w128), cblk, 0, stream>>>(
        P.P0, P.box2, P.box_b, P.A, HW, 0, P.scales, l, 0, 128);
    groupnorm_relu_pad<<<ggrid, gblk, 0, stream>>>(P.A, P.P1, P.box_g, P.box_beta, H, W);
    conv3x3_wmma<H, W><<<CG(w128), cblk, 0, stream>>>(
        P.P1, P.box2 + WSTRIDE, P.box_b + CIN, P.A, HW, 0, P.scales, l, 0, 128);
    groupnorm_relu_pad<<<ggrid, gblk, 0, stream>>>(P.A, P.P1, P.box_g + CIN, P.box_beta + CIN, H, W);
    // boxes = exp(scale^2 * (conv + bias)) fused in epilogue (mode 2)
    conv3x3_wmma<H, W><<<CG(w4), cblk, 0, stream>>>(
        P.P1, P.boxes2, P.boxes_b, P.boxes_o, PTOT, off_main, P.scales, l, 2, 4);
    conv3x3_wmma<H, W><<<CG(w4), cblk, 0, stream>>>(
        P.P1, P.conf2, P.conf_b, P.conf_o, PTOT, off_main, P.scales, l, 0, 4);
    #undef CG

    // ---- positional embedding (reads boxes just written to d_out) ----
    const int hh = H / 2, ww = W / 2;
    const int nthr = BATCH * hh * ww * 256;
    pos_embed_kernel<<<dim3((nthr + 255) / 256), gblk, 0, stream>>>(
        mask, P.boxes_o, P.pos_o, H, W, hh, ww, off_main, off_pos);
}

// ----------------------------------------------------------------------------
extern "C" void kernel_launch(void* const* d_in, const int* in_sizes, int n_in,
                              void* d_out, int out_size, void* d_ws, size_t ws_size,
                              hipStream_t stream)
{
    (void)in_sizes; (void)n_in; (void)out_size; (void)ws_size;

    static const int off13[5] = {0, 10240, 12800, 13440, 13600};
    static const int off34[5] = {0, 2560, 3200, 3360, 3400};

    Params P;
    const float* cls_w = (const float*)d_in[10];
    P.cls_b    = (const float*)d_in[11];
    P.cls_g    = (const float*)d_in[12];
    P.cls_beta = (const float*)d_in[13];
    const float* box_w = (const float*)d_in[14];
    P.box_b    = (const float*)d_in[15];
    P.box_g    = (const float*)d_in[16];
    P.box_beta = (const float*)d_in[17];
    const float* logits_w = (const float*)d_in[18];
    P.logits_b = (const float*)d_in[19];
    const float* boxes_w = (const float*)d_in[20];
    P.boxes_b  = (const float*)d_in[21];
    const float* conf_w = (const float*)d_in[22];
    P.conf_b   = (const float*)d_in[23];
    P.scales   = (const float*)d_in[24];

    float* outF = (float*)d_out;
    P.logits_o  = outF;
    P.boxes_o   = P.logits_o + (size_t)BATCH * 80 * PTOT;
    P.conf_o    = P.boxes_o  + (size_t)BATCH * 4  * PTOT;
    P.pos_o     = P.conf_o   + (size_t)BATCH * 4  * PTOT;

    // workspace layout
    float* w = (float*)d_ws;
    P.P0 = w;                 w += (size_t)BATCH * CIN * MAXCHP;
    P.P1 = w;                 w += (size_t)BATCH * CIN * MAXCHP;
    P.A  = w;                 w += (size_t)BATCH * CIN * 10240;
    P.cls2    = w;            w += (size_t)2 * CIN * CIN * 9;
    P.box2    = w;            w += (size_t)2 * CIN * CIN * 9;
    P.logits2 = w;            w += (size_t)80 * CIN * 9;
    P.boxes2  = w;            w += (size_t)4 * CIN * 9;
    P.conf2   = w;            w += (size_t)4 * CIN * 9;

    // repack all conv weights into pair-interleaved layout (once per launch)
    const dim3 rblk(256);
    #define RPK(src, dst, tot) \
        repack_weights<<<dim3(((tot) + 255) / 256), rblk, 0, stream>>>(src, dst, tot)
    RPK(cls_w,    P.cls2,    2 * CIN * CIN * 9);
    RPK(box_w,    P.box2,    2 * CIN * CIN * 9);
    RPK(logits_w, P.logits2, 80 * CIN * 9);
    RPK(boxes_w,  P.boxes2,  4 * CIN * 9);
    RPK(conf_w,   P.conf2,   4 * CIN * 9);
    #undef RPK

    for (int l = 0; l < 5; ++l) {
        const float* feat = (const float*)d_in[2 * l];
        const unsigned char* mask = (const unsigned char*)d_in[2 * l + 1];
        switch (l) {
            case 0: run_level<80, 128>(feat, mask, l, off13[l], off34[l], P, stream); break;
            case 1: run_level<40,  64>(feat, mask, l, off13[l], off34[l], P, stream); break;
            case 2: run_level<20,  32>(feat, mask, l, off13[l], off34[l], P, stream); break;
            case 3: run_level<10,  16>(feat, mask, l, off13[l], off34[l], P, stream); break;
            case 4: run_level< 5,   8>(feat, mask, l, off13[l], off34[l], P, stream); break;
        }
    }
}